// CounterFactualIPM_36756330119303
// MI455X (gfx1250) — compile-verified
//
#include <hip/hip_runtime.h>
#include <math.h>

// CounterFactual IPM / Sinkhorn-Wasserstein for MI455X (gfx1250, wave32).
// M (8192x8192 f32 distances) built once with V_WMMA_F32_16X16X4_F32, then 22
// HBM-bound streaming passes (exp recomputed on the fly). Streaming loads are
// 128-bit (global_load_b128) with non-temporal hints so the hot Sinkhorn
// vectors stay in the 192MB L2 while the 256MB matrix streams through.

typedef __attribute__((ext_vector_type(2))) float v2f;
typedef __attribute__((ext_vector_type(4))) float v4f;
typedef __attribute__((ext_vector_type(8))) float v8f;

#define NTOT   16384
#define DIM    128
#define NT     8192
#define NC     8192
#define NTILE  512            // NT/16
#define LAMBDA 10.0f
#define ITERS  10
#define ICHUNK 32             // i-chunks for the K^T @ u matvec
#define AVAL   (0.5f/8192.0f) // a_i  (i < NT)
#define BVAL   (0.5f/8192.0f) // b_j  (j < NC)

// ---- workspace layout (units: float) ----
#define SZ_M       ((size_t)NT*(size_t)NC)          // 67,108,864 floats = 256 MiB
#define OFF_U      (SZ_M)                            // u[8193]
#define OFF_S      (OFF_U + 8256)                    // s/v[8193]
#define OFF_PART   (OFF_S + 8256)                    // 32 x 8192 partials
#define OFF_ROWSUM (OFF_PART + (size_t)ICHUNK*NC)    // rowsum[8192]
#define OFF_BSUM   (OFF_ROWSUM + 8192)               // 32768 block sums
#define OFF_BMAX   (OFF_BSUM + 32768)                // 32768 block maxes
#define OFF_N2     (OFF_BMAX + 32768)                // ||x||^2 for all 16384 rows
#define OFF_SC     (OFF_N2 + 16384)                  // scalars
#define OFF_IDX    (OFF_SC + 64)                     // it[8192], ic[8192] (ints)

#define SC_EL    0   // eff_lam
#define SC_DELTA 1   // max(M)
#define SC_KD    2   // exp(-eff_lam*delta)+1e-6
#define SC_SUMU  3   // sum_{i<NT} u_i

// ---------- group partition: ordered indices where hal==1 / hal!=1 ----------
__global__ void partition_kernel(const int* __restrict__ hal,
                                 int* __restrict__ it, int* __restrict__ ic) {
  __shared__ int cnts[256];
  __shared__ int offs[256];
  int t = threadIdx.x;
  int base = t * 64;
  int c = 0;
  for (int e = 0; e < 64; ++e) c += (hal[base + e] == 1);
  cnts[t] = c;
  __syncthreads();
  if (t == 0) { int run = 0; for (int i = 0; i < 256; ++i) { offs[i] = run; run += cnts[i]; } }
  __syncthreads();
  int ones = offs[t];
  int zeros = base - ones;
  for (int e = 0; e < 64; ++e) {
    int idx = base + e;
    if (hal[idx] == 1) it[ones++] = idx; else ic[zeros++] = idx;
  }
}

// ---------- ||x_i||^2 for every row (one wave per row) ----------
__global__ void norms_kernel(const float* __restrict__ X, float* __restrict__ n2) {
  int wave = threadIdx.x >> 5, lane = threadIdx.x & 31;
  int row = blockIdx.x * 8 + wave;
  const float* xp = X + (size_t)row * DIM;
  float acc = 0.f;
#pragma unroll
  for (int q = 0; q < 4; ++q) { float v = xp[lane + 32 * q]; acc += v * v; }
  for (int off = 16; off > 0; off >>= 1) acc += __shfl_xor(acc, off, 32);
  if (lane == 0) n2[row] = acc;
}

// ---------- distance matrix via f32 WMMA: one 16x16 tile per wave ----------
// A operand per lane: Xt[m0+row][k+2*hi .. +1]; B operand: Xc[n0+row][k+2*hi .. +1]
// (ISA 16x16x4 f32 layout: lanes 0-15 hold K=0,1; lanes 16-31 hold K=2,3)
__global__ void dist_gemm_kernel(const float* __restrict__ X,
                                 const int* __restrict__ it, const int* __restrict__ ic,
                                 const float* __restrict__ n2,
                                 float* __restrict__ Mm,
                                 float* __restrict__ bsum, float* __restrict__ bmax) {
  __shared__ float sS[8], sX[8];
  int tid = threadIdx.x, wave = tid >> 5, lane = tid & 31;
  int row = lane & 15, hi = lane >> 4;
  int tile = blockIdx.x * 8 + wave;
  int ti = tile >> 9, tj = tile & 511;
  int m0 = ti << 4, n0 = tj << 4;
  const float* arow = X + (size_t)it[m0 + row] * DIM + 2 * hi;
  const float* brow = X + (size_t)ic[n0 + row] * DIM + 2 * hi;
  v8f acc = {0.f, 0.f, 0.f, 0.f, 0.f, 0.f, 0.f, 0.f};
#pragma unroll
  for (int k = 0; k < DIM; k += 4) {
    v2f a = *(const v2f*)(arow + k);
    v2f b = *(const v2f*)(brow + k);
    acc = __builtin_amdgcn_wmma_f32_16x16x4_f32(false, a, false, b, (short)0, acc,
                                                false, false);
  }
  float bn2 = n2[ic[n0 + row]];
  float lsum = 0.f, lmax = 0.f;
  int n = n0 + row;
#pragma unroll
  for (int r = 0; r < 8; ++r) {
    int m = m0 + r + 8 * hi;          // D layout: lanes 0-15 -> M=r, lanes 16-31 -> M=r+8
    float an2 = n2[it[m]];
    float sq = fmaxf(an2 - 2.0f * acc[r] + bn2, 1e-10f);
    float d = sqrtf(sq);
    __builtin_nontemporal_store(d, &Mm[(size_t)m * NC + n]);  // stream 256MB out
    lsum += d; lmax = fmaxf(lmax, d);
  }
  for (int off = 16; off > 0; off >>= 1) {
    lsum += __shfl_xor(lsum, off, 32);
    lmax = fmaxf(lmax, __shfl_xor(lmax, off, 32));
  }
  if (lane == 0) { sS[wave] = lsum; sX[wave] = lmax; }
  __syncthreads();
  if (tid == 0) {
    float s = 0.f, m = 0.f;
#pragma unroll
    for (int w = 0; w < 8; ++w) { s += sS[w]; m = fmaxf(m, sX[w]); }
    bsum[blockIdx.x] = s; bmax[blockIdx.x] = m;
  }
}

// ---------- deterministic reduction of per-block stats -> scalars ----------
__global__ void stats_kernel(const float* __restrict__ bsum, const float* __restrict__ bmax,
                             float* __restrict__ sc) {
  __shared__ float rs[256], rm[256];
  int t = threadIdx.x;
  float s = 0.f, m = 0.f;
  for (int i = t; i < 32768; i += 256) { s += bsum[i]; m = fmaxf(m, bmax[i]); }
  rs[t] = s; rm[t] = m; __syncthreads();
  for (int st = 128; st > 0; st >>= 1) {
    if (t < st) { rs[t] += rs[t + st]; rm[t] = fmaxf(rm[t], rm[t + st]); }
    __syncthreads();
  }
  if (t == 0) {
    float mean = rs[0] / (float)((double)NT * (double)NC);
    float el = LAMBDA / mean;
    sc[SC_EL] = el;
    sc[SC_DELTA] = rm[0];
    sc[SC_KD] = __expf(-el * rm[0]) + 1e-6f;
  }
}

__global__ void init_u_kernel(float* __restrict__ u) {
  int i = blockIdx.x * 256 + threadIdx.x;
  if (i < NT) u[i] = AVAL;
  else if (i == NT) u[i] = 0.5f;
}

// ---------- t = K^T u : chunked column matvec, b128 streaming ----------
// grid: (NC/1024, ICHUNK); each thread owns 4 consecutive columns.
__global__ void colmv_kernel(const float* __restrict__ Mm, const float* __restrict__ u,
                             const float* __restrict__ sc, float* __restrict__ part) {
  int j0 = (blockIdx.x * 256 + threadIdx.x) * 4;
  int i0 = blockIdx.y * 256;
  float el = sc[SC_EL];
  const v4f* Mp = (const v4f*)(Mm + (size_t)i0 * NC + j0);
  v4f acc = {0.f, 0.f, 0.f, 0.f};
#pragma unroll 2
  for (int q = 0; q < 256; ++q) {
    v4f m = __builtin_nontemporal_load(Mp);
    float uq = u[i0 + q];
    acc.x += (__expf(-el * m.x) + 1e-6f) * uq;
    acc.y += (__expf(-el * m.y) + 1e-6f) * uq;
    acc.z += (__expf(-el * m.z) + 1e-6f) * uq;
    acc.w += (__expf(-el * m.w) + 1e-6f) * uq;
    Mp += NC / 4;
  }
  *(v4f*)(part + (size_t)blockIdx.y * NC + j0) = acc;
}

__global__ void sumu_kernel(const float* __restrict__ u, float* __restrict__ sc) {
  __shared__ float rs[256];
  int t = threadIdx.x; float s = 0.f;
  for (int i = t; i < NT; i += 256) s += u[i];
  rs[t] = s; __syncthreads();
  for (int st = 128; st > 0; st >>= 1) { if (t < st) rs[t] += rs[t + st]; __syncthreads(); }
  if (t == 0) sc[SC_SUMU] = rs[0];
}

// s_j = b_j / t_j  (incl. slack row in t, and slack element s[NC]); 4 cols/thread
__global__ void finish_t_kernel(const float* __restrict__ part, const float* __restrict__ u,
                                const float* __restrict__ sc, float* __restrict__ s) {
  int j0 = (blockIdx.x * 256 + threadIdx.x) * 4;
  float Kd = sc[SC_KD];
  float uNT = u[NT];
  v4f t = {Kd * uNT, Kd * uNT, Kd * uNT, Kd * uNT};
#pragma unroll
  for (int c = 0; c < ICHUNK; ++c) {
    v4f p = *(const v4f*)(part + (size_t)c * NC + j0);
    t.x += p.x; t.y += p.y; t.z += p.z; t.w += p.w;
  }
  v4f sv = {BVAL / t.x, BVAL / t.y, BVAL / t.z, BVAL / t.w};
  *(v4f*)(s + j0) = sv;
  if (blockIdx.x == 0 && threadIdx.x == 0) {
    float tS = Kd * sc[SC_SUMU] + (1.0f + 1e-6f) * uNT;
    s[NC] = 0.5f / tS;
  }
}

// u[NT] = a_slack / (K-slack-row @ s)
__global__ void slack_u_kernel(const float* __restrict__ s, const float* __restrict__ sc,
                               float* __restrict__ u) {
  __shared__ float rs[256];
  int t = threadIdx.x; float a = 0.f;
  for (int j = t; j < NC; j += 256) a += s[j];
  rs[t] = a; __syncthreads();
  for (int st = 128; st > 0; st >>= 1) { if (t < st) rs[t] += rs[t + st]; __syncthreads(); }
  if (t == 0) {
    float Kd = sc[SC_KD];
    u[NT] = 0.5f / (Kd * rs[0] + (1.0f + 1e-6f) * s[NC]);
  }
}

// u_i = a_i / (K @ s)_i : one block per row, b128 streaming, LDS tree reduce
__global__ void rowmv_update_kernel(const float* __restrict__ Mm, const float* __restrict__ s,
                                    const float* __restrict__ sc, float* __restrict__ u) {
  __shared__ float rs[256];
  int t = threadIdx.x; int i = blockIdx.x;
  float el = sc[SC_EL];
  const v4f* Mp = (const v4f*)(Mm + (size_t)i * NC);
  const v4f* sp = (const v4f*)s;
  float acc = 0.f;
#pragma unroll
  for (int j = t; j < NC / 4; j += 256) {
    v4f m = __builtin_nontemporal_load(Mp + j);
    v4f sv = sp[j];
    acc += (__expf(-el * m.x) + 1e-6f) * sv.x
         + (__expf(-el * m.y) + 1e-6f) * sv.y
         + (__expf(-el * m.z) + 1e-6f) * sv.z
         + (__expf(-el * m.w) + 1e-6f) * sv.w;
  }
  rs[t] = acc; __syncthreads();
  for (int st = 128; st > 0; st >>= 1) { if (t < st) rs[t] += rs[t + st]; __syncthreads(); }
  if (t == 0) {
    float Kd = sc[SC_KD];
    u[i] = AVAL / (rs[0] + Kd * s[NC]);
  }
}

// rowsum_i = u_i * ( sum_j v_j K_ij M_ij  +  v[NC]*Kd*delta )
__global__ void final_row_kernel(const float* __restrict__ Mm, const float* __restrict__ s,
                                 const float* __restrict__ u, const float* __restrict__ sc,
                                 float* __restrict__ rowsum) {
  __shared__ float rs[256];
  int t = threadIdx.x; int i = blockIdx.x;
  float el = sc[SC_EL];
  const v4f* Mp = (const v4f*)(Mm + (size_t)i * NC);
  const v4f* sp = (const v4f*)s;
  float acc = 0.f;
#pragma unroll
  for (int j = t; j < NC / 4; j += 256) {
    v4f m = __builtin_nontemporal_load(Mp + j);
    v4f sv = sp[j];
    acc += (__expf(-el * m.x) + 1e-6f) * m.x * sv.x
         + (__expf(-el * m.y) + 1e-6f) * m.y * sv.y
         + (__expf(-el * m.z) + 1e-6f) * m.z * sv.z
         + (__expf(-el * m.w) + 1e-6f) * m.w * sv.w;
  }
  rs[t] = acc; __syncthreads();
  for (int st = 128; st > 0; st >>= 1) { if (t < st) rs[t] += rs[t + st]; __syncthreads(); }
  if (t == 0) {
    float Kd = sc[SC_KD], delta = sc[SC_DELTA];
    rowsum[i] = u[i] * (rs[0] + Kd * delta * s[NC]);
  }
}

// out = 2 * ( sum_i rowsum_i + u[NT]*Kd*delta*sum_j v_j )   (corner Mt==0)
__global__ void final_reduce_kernel(const float* __restrict__ rowsum, const float* __restrict__ s,
                                    const float* __restrict__ u, const float* __restrict__ sc,
                                    float* __restrict__ out) {
  __shared__ float r1[256], r2[256];
  int t = threadIdx.x;
  float a = 0.f, b = 0.f;
  for (int i = t; i < NT; i += 256) a += rowsum[i];
  for (int j = t; j < NC; j += 256) b += s[j];
  r1[t] = a; r2[t] = b; __syncthreads();
  for (int st = 128; st > 0; st >>= 1) {
    if (t < st) { r1[t] += r1[t + st]; r2[t] += r2[t + st]; }
    __syncthreads();
  }
  if (t == 0) {
    float Kd = sc[SC_KD], delta = sc[SC_DELTA];
    out[0] = 2.0f * (r1[0] + u[NT] * r2[0] * Kd * delta);  // IPM_ALPHA == 1
  }
}

extern "C" void kernel_launch(void* const* d_in, const int* in_sizes, int n_in,
                              void* d_out, int out_size, void* d_ws, size_t ws_size,
                              hipStream_t stream) {
  const float* X = (const float*)d_in[0];
  const int* hal = (const int*)d_in[1];
  float* ws     = (float*)d_ws;
  float* Mm     = ws;                 // 256 MiB distance matrix
  float* u      = ws + OFF_U;
  float* s      = ws + OFF_S;         // s during iterations; v at the end
  float* part   = ws + OFF_PART;
  float* rowsum = ws + OFF_ROWSUM;
  float* bsum   = ws + OFF_BSUM;
  float* bmax   = ws + OFF_BMAX;
  float* n2     = ws + OFF_N2;
  float* sc     = ws + OFF_SC;
  int*   it     = (int*)(ws + OFF_IDX);
  int*   ic     = it + NT;
  float* out    = (float*)d_out;

  partition_kernel<<<1, 256, 0, stream>>>(hal, it, ic);
  norms_kernel<<<NTOT / 8, 256, 0, stream>>>(X, n2);
  dist_gemm_kernel<<<(NTILE * NTILE) / 8, 256, 0, stream>>>(X, it, ic, n2, Mm, bsum, bmax);
  stats_kernel<<<1, 256, 0, stream>>>(bsum, bmax, sc);
  init_u_kernel<<<(NT + 256 + 255) / 256, 256, 0, stream>>>(u);

  for (int itn = 0; itn < ITERS; ++itn) {
    colmv_kernel<<<dim3(NC / 1024, ICHUNK), 256, 0, stream>>>(Mm, u, sc, part);
    sumu_kernel<<<1, 256, 0, stream>>>(u, sc);
    finish_t_kernel<<<NC / 1024, 256, 0, stream>>>(part, u, sc, s);
    slack_u_kernel<<<1, 256, 0, stream>>>(s, sc, u);
    rowmv_update_kernel<<<NT, 256, 0, stream>>>(Mm, s, sc, u);
  }
  // v = b / (K^T u)  — one more column matvec, result lands in s (== v)
  colmv_kernel<<<dim3(NC / 1024, ICHUNK), 256, 0, stream>>>(Mm, u, sc, part);
  sumu_kernel<<<1, 256, 0, stream>>>(u, sc);
  finish_t_kernel<<<NC / 1024, 256, 0, stream>>>(part, u, sc, s);

  final_row_kernel<<<NT, 256, 0, stream>>>(Mm, s, u, sc, rowsum);
  final_reduce_kernel<<<1, 256, 0, stream>>>(rowsum, s, u, sc, out);
}